// Encoder_86105504350850
// MI455X (gfx1250) — compile-verified
//
#include <hip/hip_runtime.h>

// ---------------------------------------------------------------------------
// The entire reference network is affine (no activations). The returned
// output depends only on the flights slice:  y = x @ Weff(16x16) + beff(16)
// per flight vector, where Weff = M0*F1*F2*F3*DF1*DF2*DF3*Mdec.
//
// Roofline: 1,048,576 vectors * 512 FLOP = 0.54 GFLOP over 134 MB traffic
// -> 4 FLOP/byte, HBM-bound (~5.8 us @ 23.3 TB/s). We therefore:
//   Kernel 1 (1 block): fold all params into Weff/beff in d_ws.
//   Kernel 2: streaming v_wmma_f32_16x16x4_f32 (native f32 matrix op),
//             one 16x16 tile per wave iteration, bias preloaded into C.
// ---------------------------------------------------------------------------

typedef __attribute__((ext_vector_type(2))) float v2f;
typedef __attribute__((ext_vector_type(8))) float v8f;

#define STATE_STRIDE 1684   // FCS*NF + STS*NT + STS = 1024 + 640 + 20
#define TOTAL_TILES  65536  // (16384 * 64) / 16
#define TILES_PER_WAVE 8

struct EncParams {
    const float *airW, *airB, *ftW, *ftB, *etaW, *etaB, *timW, *timB;
    const float *f1W, *f1B, *f2W, *f2B, *f3W, *f3B;
    const float *df1W, *df1B, *df2W, *df2B, *df3W, *df3B;
    const float *airDW, *airDB, *ftDW, *ftDB, *etaDW, *etaDB, *timDW, *timDB;
};

// One chained affine step in LDS: Aout[16 x dout] = Ain[16 x din] @ W,
// bout = bin @ W + Wb.  Small matrices; one workgroup, scalar math is fine.
__device__ inline void mm_step(const float* Ain, const float* W,
                               const float* bin, const float* Wb,
                               float* Aout, float* bout, int din, int dout)
{
    for (int idx = threadIdx.x; idx < 16 * dout; idx += blockDim.x) {
        int r = idx / dout, c = idx - r * dout;
        float s = 0.f;
        for (int k = 0; k < din; ++k) s += Ain[r * din + k] * W[k * dout + c];
        Aout[idx] = s;
    }
    for (int c = threadIdx.x; c < dout; c += blockDim.x) {
        float s = Wb[c];
        for (int k = 0; k < din; ++k) s += bin[k] * W[k * dout + c];
        bout[c] = s;
    }
    __syncthreads();
}

// Kernel 1: fold the whole affine chain into Weff (ws[0..255], row-major
// [k][n]) and beff (ws[256..271]).
__global__ void __launch_bounds__(256) build_effective(EncParams P, float* ws)
{
    __shared__ float A[16 * 128];
    __shared__ float Bf[16 * 128];
    __shared__ float Md[16 * 16];
    __shared__ float bv[128], bv2[128], bdec[16];
    const int tid = threadIdx.x;

    for (int i = tid; i < 256; i += blockDim.x) { A[i] = 0.f; Md[i] = 0.f; }
    __syncthreads();

    // M0: encoder conv block (f = [air | ftp | tim | eta])
    if (tid < 16)              { int r = tid >> 2, c = tid & 3; A[r * 16 + c] = P.airW[tid]; }
    if (tid < 40)              { int r = tid >> 2, c = tid & 3; A[(4 + r) * 16 + (4 + c)] = P.ftW[tid]; }
    if (tid >= 64 && tid < 68) A[14 * 16 + 8  + (tid - 64)] = P.timW[tid - 64];
    if (tid >= 68 && tid < 72) A[15 * 16 + 12 + (tid - 68)] = P.etaW[tid - 68];
    if (tid >= 72 && tid < 76) bv[      tid - 72] = P.airB[tid - 72];
    if (tid >= 76 && tid < 80) bv[4  + (tid - 76)] = P.ftB[tid - 76];
    if (tid >= 80 && tid < 84) bv[8  + (tid - 80)] = P.timB[tid - 80];
    if (tid >= 84 && tid < 88) bv[12 + (tid - 84)] = P.etaB[tid - 84];
    // Mdec: decoder block (out = [airDe | ftDe | timeDe | etaDe])
    if (tid >= 96  && tid < 112) { int q = tid - 96;  int r = q >> 2, c = q & 3; Md[r * 16 + c] = P.airDW[q]; }
    if (tid >= 112 && tid < 152) { int q = tid - 112; int r = q / 10, c = q % 10; Md[(4 + r) * 16 + 4 + c] = P.ftDW[q]; }
    if (tid >= 152 && tid < 156) Md[(8  + (tid - 152)) * 16 + 14] = P.timDW[tid - 152];
    if (tid >= 156 && tid < 160) Md[(12 + (tid - 156)) * 16 + 15] = P.etaDW[tid - 156];
    if (tid >= 160 && tid < 164) bdec[      tid - 160] = P.airDB[tid - 160];
    if (tid >= 164 && tid < 174) bdec[4 + (tid - 164)] = P.ftDB[tid - 164];
    if (tid == 174) bdec[14] = P.timDB[0];
    if (tid == 175) bdec[15] = P.etaDB[0];
    __syncthreads();

    mm_step(A,  P.f1W,  bv,  P.f1B,  Bf, bv2, 16, 64);
    mm_step(Bf, P.f2W,  bv2, P.f2B,  A,  bv,  64, 128);
    mm_step(A,  P.f3W,  bv,  P.f3B,  Bf, bv2, 128, 16);
    mm_step(Bf, P.df1W, bv2, P.df1B, A,  bv,  16, 128);
    mm_step(A,  P.df2W, bv,  P.df2B, Bf, bv2, 128, 64);
    mm_step(Bf, P.df3W, bv2, P.df3B, A,  bv,  64, 16);

    // Weff = A @ Mdec ; beff = bv @ Mdec + bdec
    for (int idx = tid; idx < 256; idx += blockDim.x) {
        int r = idx >> 4, c = idx & 15;
        float s = 0.f;
        for (int k = 0; k < 16; ++k) s += A[r * 16 + k] * Md[k * 16 + c];
        ws[idx] = s;
    }
    if (tid < 16) {
        float s = bdec[tid];
        for (int k = 0; k < 16; ++k) s += bv[k] * Md[k * 16 + tid];
        ws[256 + tid] = s;
    }
}

// Kernel 2: streaming y = x @ Weff + beff with V_WMMA_F32_16X16X4_F32.
// One 16x16 tile (16 flight vectors) per wave per iteration; K=16 via four
// chained K=4 WMMAs. Tiles never straddle a batch row (64 flights = 4 tiles).
__global__ void __launch_bounds__(256) encoder_fused(
    const float* __restrict__ state,
    const float* __restrict__ wb,     // ws: Weff[256] row-major [k][n], beff[16]
    float* __restrict__ out)
{
    const int lane = threadIdx.x & 31;
    const int wave = (blockIdx.x * blockDim.x + threadIdx.x) >> 5;
    const int half = lane >> 4;    // 0: K%4 in {0,1} / rows 0..7 of D
    const int n    = lane & 15;    // column / row-within-half

    // B operand (Weff), ISA 16x16x4 f32 layout: lanes0-15 hold K={0,1},
    // lanes16-31 hold K={2,3} across the 2 VGPRs of each K-chunk.
    v2f bmat[4];
#pragma unroll
    for (int kc = 0; kc < 4; ++kc) {
        const int k0 = 4 * kc + 2 * half;
        bmat[kc].x = wb[(k0 + 0) * 16 + n];
        bmat[kc].y = wb[(k0 + 1) * 16 + n];
    }
    const float bias = wb[256 + n];

    int t = wave * TILES_PER_WAVE;
    for (int i = 0; i < TILES_PER_WAVE; ++i, ++t) {
        // tile t -> batch row t>>2, flight block t&3 (1KB contiguous)
        const float* src  = state + (size_t)(t >> 2) * STATE_STRIDE
                                  + (size_t)(t & 3) * 256;
        const float* rowp = src + n * 16 + 2 * half;   // 8B-aligned
        v2f amat[4];
#pragma unroll
        for (int kc = 0; kc < 4; ++kc)
            amat[kc] = __builtin_nontemporal_load((const v2f*)(rowp + 4 * kc));

        v8f c;
#pragma unroll
        for (int v = 0; v < 8; ++v) c[v] = bias;   // bias preloaded into C

#pragma unroll
        for (int kc = 0; kc < 4; ++kc)
            c = __builtin_amdgcn_wmma_f32_16x16x4_f32(
                    false, amat[kc], false, bmat[kc],
                    (short)0, c, false, false);

        // D layout: VGPR v -> row v (lanes 0-15), row v+8 (lanes 16-31)
        float* dst = out + (size_t)t * 256;
#pragma unroll
        for (int v = 0; v < 8; ++v)
            __builtin_nontemporal_store(c[v], &dst[(v + 8 * half) * 16 + n]);
    }
}

extern "C" void kernel_launch(void* const* d_in, const int* in_sizes, int n_in,
                              void* d_out, int out_size, void* d_ws, size_t ws_size,
                              hipStream_t stream)
{
    (void)in_sizes; (void)n_in; (void)out_size; (void)ws_size;
    const float* state = (const float*)d_in[0];
    // params flattened in _DIMS insertion order: entry i -> W at 1+2i, b at 2+2i
    auto W = [&](int i) { return (const float*)d_in[1 + 2 * i]; };
    auto Bb = [&](int i) { return (const float*)d_in[2 + 2 * i]; };

    EncParams P;
    P.airW = W(0);  P.airB = Bb(0);   // airConv
    P.ftW  = W(1);  P.ftB  = Bb(1);   // flightTypeConv
    P.etaW = W(2);  P.etaB = Bb(2);   // etaConv
    P.timW = W(3);  P.timB = Bb(3);   // timeConv
    P.f1W  = W(4);  P.f1B  = Bb(4);
    P.f2W  = W(5);  P.f2B  = Bb(5);
    P.f3W  = W(6);  P.f3B  = Bb(6);
    // t1..t4 (7..10), c1..c4 (11..14) are dead code in the reference
    P.df1W = W(15); P.df1B = Bb(15);
    P.df2W = W(16); P.df2B = Bb(16);
    P.df3W = W(17); P.df3B = Bb(17);
    P.airDW = W(18); P.airDB = Bb(18);
    P.ftDW  = W(19); P.ftDB  = Bb(19);
    P.etaDW = W(20); P.etaDB = Bb(20);
    P.timDW = W(21); P.timDB = Bb(21);

    float* ws = (float*)d_ws;
    build_effective<<<1, 256, 0, stream>>>(P, ws);

    // 65536 tiles / 8 per wave = 8192 waves; 8 wave32s per 256-thread block
    const int blocks = (TOTAL_TILES / TILES_PER_WAVE) / 8;   // 1024
    encoder_fused<<<blocks, 256, 0, stream>>>(state, ws, (float*)d_out);
}